// EquiConv_6897717477396
// MI455X (gfx1250) — compile-verified
//
#include <hip/hip_runtime.h>
#include <math.h>
#include <stdint.h>

typedef __attribute__((ext_vector_type(2))) float v2f;
typedef __attribute__((ext_vector_type(8))) float v8f;

#define BM      128                  // rows per block
#define NW      8                    // waves per block (wave32)
#define RPW     16                   // rows per wave tile (WMMA M)
#define FC      64                   // K dimension of both GEMMs
#define WN      2304                 // columns of W2
#define SLABW   96                   // W2 slab width staged in LDS
#define SLABP   104                  // padded LDS row stride (floats): +16 banks per 2 rows
#define NSLAB   (WN / SLABW)         // 24
#define CPS     (SLABW / 16)         // 6 chunks of 16 cols per slab
#define CSTRIDE 132                  // per-row coefficient stride (floats)

#define SLABBYTES   (64 * SLABP * 4) // 26624 B per slab buffer
#define HBYTES      (NW * RPW * 64 * 4)      // 32768
#define COEFBYTES   (BM * CSTRIDE * 4)       // 67584
// smem: [slab0 (overlaid by pair-packed W1 in phase 1)][slab1][sH][sCoef] = 150 KB

__global__ __launch_bounds__(256)
void equiconv_fused_kernel(const float* __restrict__ fea_in1,   // (E,80)
                           const float* __restrict__ fea_in2,   // (E,4)
                           const float* __restrict__ fea_weight,// (E,64)
                           const float* __restrict__ W1,        // (64,64)
                           const float* __restrict__ W2,        // (64,2304)
                           float* __restrict__ out)             // (E,80)
{
    __shared__ __align__(16) char smem[2 * SLABBYTES + HBYTES + COEFBYTES];

    v2f*   sW1   = (v2f*)smem;                              // phase-1 overlay on slab buf 0
    float* sH    = (float*)(smem + 2 * SLABBYTES);
    float* sCoef = (float*)(smem + 2 * SLABBYTES + HBYTES);

    const int tid  = threadIdx.x;
    const int wave = tid >> 5;
    const int lane = tid & 31;
    const int lo   = lane & 15;       // N (or M) index within 16
    const int hi   = lane >> 4;       // selects K pair {0,1} vs {2,3} of each quad
    const int blockRow = blockIdx.x * BM;

    // ---------------- stage W1*0.125 as K-pairs (phase-1 B layout) ------
    for (int q = tid; q < 32 * 64; q += 256) {
        const int p = q >> 6, j = q & 63;
        v2f v;
        v.x = 0.125f * W1[(2 * p + 0) * 64 + j];
        v.y = 0.125f * W1[(2 * p + 1) * 64 + j];
        sW1[q] = v;
    }

    // ---------------- per-row contraction coefficients ------------------
    {
        const float k00  = 0.125f * 0.125f;                            // 1/8*sqrt(1/64)
        const float k110 = 0.125f * 0.17677669529f * 0.57735026919f;   // 1/8*sqrt(1/32)/sqrt3
        const float k011 = 0.125f * 0.21650635095f * 0.57735026919f;   // 1/8*sqrt(3/64)/sqrt3
        const float k101 = 0.125f * 0.30618621785f * 0.57735026919f;   // 1/8*sqrt(3/32)/sqrt3
        if (tid < BM) {
            const int e = blockRow + tid;
            const float* f1 = fea_in1 + (size_t)e * 80;
            const float* f2 = fea_in2 + (size_t)e * 4;
            const float x20 = f2[0];
            const float y0 = f2[1], y1 = f2[2], y2 = f2[3];
            float* cf = sCoef + tid * CSTRIDE;
            #pragma unroll
            for (int u = 0; u < 32; ++u) {
                const float x = f1[u];
                cf[u]      = k00  * x20 * x;
                cf[48 + u] = k011 * x;
            }
            #pragma unroll
            for (int u = 0; u < 16; ++u) {
                const float a = f1[32 + 3 * u + 0];
                const float b = f1[32 + 3 * u + 1];
                const float c = f1[32 + 3 * u + 2];
                cf[32 + u]         = k110 * (a * y0 + b * y1 + c * y2);
                cf[80 + 3 * u + 0] = k101 * x20 * a;
                cf[80 + 3 * u + 1] = k101 * x20 * b;
                cf[80 + 3 * u + 2] = k101 * x20 * c;
            }
            cf[128] = y0; cf[129] = y1; cf[130] = y2;
        }
    }
    __syncthreads();

    // ---------------- phase 1: h = 1.679*silu(fw @ W1/8) ----------------
    v2f fa[16];
    {
        const float* fw = fea_weight + (size_t)(blockRow + wave * RPW + lo) * FC + 2 * hi;
        #pragma unroll
        for (int kk = 0; kk < 16; ++kk)
            fa[kk] = *(const v2f*)(fw + 4 * kk);
    }
    float* myH = sH + wave * (RPW * 64);
    #pragma unroll
    for (int n = 0; n < 4; ++n) {
        v8f acc = {0.f,0.f,0.f,0.f,0.f,0.f,0.f,0.f};
        #pragma unroll
        for (int kk = 0; kk < 16; ++kk) {
            const v2f b = sW1[(2 * kk + hi) * 64 + n * 16 + lo];
            acc = __builtin_amdgcn_wmma_f32_16x16x4_f32(false, fa[kk], false, b,
                                                        (short)0, acc, false, false);
        }
        #pragma unroll
        for (int i = 0; i < 8; ++i) {   // C layout: VGPR i -> row i+8*hi, col n*16+lo
            const float z = acc[i];
            myH[(i + 8 * hi) * 64 + n * 16 + lo] = 1.679f * z / (1.0f + __expf(-z));
        }
    }
    v2f ha[16];     // re-fragment H (C layout -> A layout) through LDS
    {
        const float* hp = myH + lo * 64 + 2 * hi;
        #pragma unroll
        for (int kk = 0; kk < 16; ++kk)
            ha[kk] = *(const v2f*)(hp + 4 * kk);
    }
    __syncthreads();   // sW1 overlay region may now be reused as async slab buf 0

    // ---------------- phase 2: fused  w = h@W2/8  + contractions --------
    // Async (ASYNCcnt-tracked) memory->LDS staging of W2 slabs, double-buffered.
    auto stage_async = [&](int slab, int bufsel) {
        const uint32_t ldsBase = (uint32_t)(uintptr_t)smem + (uint32_t)(bufsel * SLABBYTES);
        const float* gbase = W2 + (size_t)slab * SLABW;
        #pragma unroll
        for (int it = 0; it < 6; ++it) {          // 1536 b128 transfers / 256 threads
            const int i   = tid + it * 256;
            const int r   = i / 24;               // W2 row 0..63
            const int seg = i - r * 24;           // 16-byte segment within 96-col row
            const uint32_t dst = ldsBase + (uint32_t)(r * (SLABP * 4) + seg * 16);
            const float* src = gbase + (size_t)r * WN + seg * 4;
            asm volatile("global_load_async_to_lds_b128 %0, %1, off"
                         :: "v"(dst), "v"(src) : "memory");
        }
    };

    v8f acc00a = {0.f,0.f,0.f,0.f,0.f,0.f,0.f,0.f};   // out0 cols  0..15
    v8f acc00b = {0.f,0.f,0.f,0.f,0.f,0.f,0.f,0.f};   // out0 cols 16..31
    v8f accT   = {0.f,0.f,0.f,0.f,0.f,0.f,0.f,0.f};   // t011
    v8f accK0  = {0.f,0.f,0.f,0.f,0.f,0.f,0.f,0.f};   // t101 k=0
    v8f accK1  = {0.f,0.f,0.f,0.f,0.f,0.f,0.f,0.f};   // t101 k=1
    v8f accK2  = {0.f,0.f,0.f,0.f,0.f,0.f,0.f,0.f};   // t101 k=2
    const float* cfBase = sCoef + (wave * RPW + 8 * hi) * CSTRIDE;

    stage_async(0, 0);
    stage_async(1, 1);

    for (int slab = 0; slab < NSLAB; ++slab) {
        // in-order async completion: <=6 outstanding means slab's stage is done
        if (slab + 1 < NSLAB) asm volatile("s_wait_asynccnt 0x6" ::: "memory");
        else                  asm volatile("s_wait_asynccnt 0x0" ::: "memory");
        __syncthreads();                           // slab data visible to all waves

        const float* buf = (const float*)(smem + (slab & 1) * SLABBYTES);
        for (int ci = 0; ci < CPS; ++ci) {
            const int c = slab * CPS + ci;         // global 16-col chunk index
            v8f wt = {0.f,0.f,0.f,0.f,0.f,0.f,0.f,0.f};
            const float* base = buf + ci * 16 + lo;
            #pragma unroll
            for (int kk = 0; kk < 16; ++kk) {
                v2f b;
                b.x = base[(4 * kk + 2 * hi + 0) * SLABP];
                b.y = base[(4 * kk + 2 * hi + 1) * SLABP];
                wt = __builtin_amdgcn_wmma_f32_16x16x4_f32(false, ha[kk], false, b,
                                                           (short)0, wt, false, false);
            }
            if (c < 96) {                          // w00 / w110 regions -> out0
                const int idx = (c < 64) ? (c >> 1) : (32 + ((c - 64) >> 1));
                v8f& dst = (c & 1) ? acc00b : acc00a;
                #pragma unroll
                for (int i = 0; i < 8; ++i)
                    dst[i] = fmaf(cfBase[i * CSTRIDE + idx], wt[i], dst[i]);
            } else if (c < 128) {                  // w011 region -> t011
                const int idx = 48 + (c - 96);
                #pragma unroll
                for (int i = 0; i < 8; ++i)
                    accT[i] = fmaf(cfBase[i * CSTRIDE + idx], wt[i], accT[i]);
            } else {                               // w101 region -> t101[k]
                const int idx = 80 + 3 * (c - 128);
                #pragma unroll
                for (int i = 0; i < 8; ++i) {
                    const float* cp = cfBase + i * CSTRIDE + idx;
                    accK0[i] = fmaf(cp[0], wt[i], accK0[i]);
                    accK1[i] = fmaf(cp[1], wt[i], accK1[i]);
                    accK2[i] = fmaf(cp[2], wt[i], accK2[i]);
                }
            }
        }
        __syncthreads();                           // all readers done with buf
        if (slab + 2 < NSLAB) stage_async(slab + 2, slab & 1);
    }

    // ---------------- epilogue: out = [out0 | out1.reshape(48)] ---------
    float* outw = out + (size_t)(blockRow + wave * RPW) * 80;
    #pragma unroll
    for (int i = 0; i < 8; ++i) {
        float* o = outw + (size_t)(i + 8 * hi) * 80;
        const float* cf = cfBase + i * CSTRIDE;
        o[lo]      = acc00a[i];
        o[16 + lo] = acc00b[i];
        const float t = accT[i];
        o[32 + 3 * lo + 0] = fmaf(t, cf[128], accK0[i]);
        o[32 + 3 * lo + 1] = fmaf(t, cf[129], accK1[i]);
        o[32 + 3 * lo + 2] = fmaf(t, cf[130], accK2[i]);
    }
}

extern "C" void kernel_launch(void* const* d_in, const int* in_sizes, int n_in,
                              void* d_out, int out_size, void* d_ws, size_t ws_size,
                              hipStream_t stream) {
    (void)n_in; (void)out_size; (void)d_ws; (void)ws_size;
    const float* fea_in1    = (const float*)d_in[0];
    const float* fea_in2    = (const float*)d_in[1];
    const float* fea_weight = (const float*)d_in[2];
    const float* W1         = (const float*)d_in[3];
    const float* W2         = (const float*)d_in[4];
    float* outp             = (float*)d_out;

    const int E = in_sizes[2] / FC;      // 65536
    dim3 grid(E / BM), block(256);
    hipLaunchKernelGGL(equiconv_fused_kernel, grid, block, 0, stream,
                       fea_in1, fea_in2, fea_weight, W1, W2, outp);
}